// DiffOmegaVectorNorm_38740605010342
// MI455X (gfx1250) — compile-verified
//
#include <hip/hip_runtime.h>
#include <stdint.h>

// Problem geometry (fixed by the reference): B=2, C=4, D=64, H=256, W=256.
// Element strides: channel = 1<<22, z-plane = 1<<16, row = 1<<8.
//
// out = sum_{interior} w * |omega_p - omega_t| / sum_{interior} w
//   w(z,y,x)   = 1 iff all 27 mask voxels in the 3x3x3 neighborhood are 1
//   omega uses raw differences (scale 10 and 1/(2*5) cancel):
//     ox = (dW[y+1]-dW[y-1]) - (dV[z+1]-dV[z-1])
//     oy = (dU[z+1]-dU[z-1]) - (dW[x+1]-dW[x-1])
//     oz = (dV[x+1]-dV[x-1]) - (dU[y+1]-dU[y-1])
//   dU,dV,dW = predicts - targets for channels 1,2,3.

#define TXI   64            // interior columns per tile
#define TYI   16            // interior rows per tile
#define ROWS  (TYI + 2)     // 18 (y halo)
#define COLS  (TXI + 2)     // 66 (x halo)
#define PLANE (ROWS * COLS) // 1188 floats per staged mask plane
#define ZCHUNK 8
#define NBLK  1024          // 4 x-tiles * 16 y-tiles * 8 z-chunks * 2 batches

__global__ __launch_bounds__(256) void vortdiff_kernel(
    const float* __restrict__ P, const float* __restrict__ T,
    const float* __restrict__ M, float* __restrict__ ws)
{
  __shared__ float smask[4 * PLANE];   // 4-slot rolling z-plane buffer (19 KB)
  __shared__ float red[512];

  const int tid = threadIdx.x;
  const int bid = blockIdx.x;
  const int tx = bid & 3;          // 4 x-tiles
  const int ty = (bid >> 2) & 15;  // 16 y-tiles
  const int zc = (bid >> 6) & 7;   // 8 z-chunks
  const int b  = (bid >> 9) & 1;   // 2 batches

  const int x0 = 1 + tx * TXI;
  const int y0 = 1 + ty * TYI;
  const int zStart = 1 + zc * ZCHUNK;
  const int zEnd   = min(62, zStart + ZCHUNK - 1);

  // ---- async-copy plan: 5 B32 issues/wave per mask plane (uniform count) ----
  int goff[5];   // byte offset within a z-plane of the mask tensor (clamped)
  int eoff[5];   // element index within the staged LDS plane
  #pragma unroll
  for (int k = 0; k < 5; ++k) {
    int e = tid + 256 * k;
    if (e > PLANE - 1) e = PLANE - 1;          // duplicate tail: keeps per-wave
    eoff[k] = e;                               // async issue count uniform (=5)
    int r = e / COLS, c = e - r * COLS;
    int gy = y0 - 1 + r; if (gy > 255) gy = 255;   // clamped halo cells are
    int gx = x0 - 1 + c; if (gx > 255) gx = 255;   // never read by valid points
    goff[k] = ((gy << 8) + gx) * 4;
  }
  const float* Mb = M + ((size_t)b << 22);
  const unsigned lds0 = (unsigned)(uintptr_t)&smask[0];

  auto issue_plane = [&](int pz) {
    const float* sb = Mb + ((size_t)pz << 16);     // uniform -> SGPR pair
    const int slotbase = (pz & 3) * PLANE;
    #pragma unroll
    for (int k = 0; k < 5; ++k) {
      unsigned la = lds0 + (unsigned)((slotbase + eoff[k]) * 4);
      int vo = goff[k];
      // CDNA5 async global->LDS copy (GVS mode), tracked by ASYNCcnt.
      asm volatile("global_load_async_to_lds_b32 %0, %1, %2"
                   :: "v"(la), "v"(vo), "s"(sb) : "memory");
    }
  };

  // Preload planes zStart-1, zStart, zStart+1 (slots (z-1..z+1)&3).
  issue_plane(zStart - 1);
  issue_plane(zStart);
  issue_plane(zStart + 1);

  const int lx  = tid & 63;
  const int ly0 = tid >> 6;
  const int gx  = x0 + lx;
  const size_t cb = ((size_t)(b * 4) << 22);
  const float* up = P + cb + ((size_t)1 << 22);
  const float* vp = P + cb + ((size_t)2 << 22);
  const float* wp = P + cb + ((size_t)3 << 22);
  const float* ut = T + cb + ((size_t)1 << 22);
  const float* vt = T + cb + ((size_t)2 << 22);
  const float* wt = T + cb + ((size_t)3 << 22);

  float accN = 0.f, accC = 0.f;

  for (int z = zStart; z <= zEnd; ++z) {
    // Prefetch plane z+2 into slot (z+2)&3 (disjoint from the 3 slots read
    // below), then drain everything except that in-flight batch.
    if (z + 1 <= zEnd) {
      issue_plane(z + 2);
      asm volatile("s_wait_asynccnt 0x5" ::: "memory");
    } else {
      asm volatile("s_wait_asynccnt 0x0" ::: "memory");
    }
    __syncthreads();   // make every wave's copies visible to all

    const int s0 = ((z - 1) & 3) * PLANE;
    const int s1 = ((z    ) & 3) * PLANE;
    const int s2 = ((z + 1) & 3) * PLANE;

    #pragma unroll
    for (int j = 0; j < 4; ++j) {
      const int ly = ly0 + 4 * j;
      const int gy = y0 + ly;
      if (gy <= 254 && gx <= 254) {
        const int rr = (ly + 1) * COLS + (lx + 1);
        // w = product of the 27-neighborhood of the (0/1) mask
        float wprod = 1.f;
        #pragma unroll
        for (int dr = -1; dr <= 1; ++dr)
          #pragma unroll
          for (int dc = -1; dc <= 1; ++dc) {
            const int o = rr + dr * COLS + dc;
            wprod *= smask[s0 + o] * smask[s1 + o] * smask[s2 + o];
          }
        if (wprod != 0.f) {   // ~5.9% of points: skip 24 global loads otherwise
          const int i0  = (z << 16) + (gy << 8) + gx;
          const int izp = i0 + 65536, izm = i0 - 65536;
          const int iyp = i0 + 256,   iym = i0 - 256;
          const int ixp = i0 + 1,     ixm = i0 - 1;
          const float ox = ((wp[iyp] - wt[iyp]) - (wp[iym] - wt[iym]))
                         - ((vp[izp] - vt[izp]) - (vp[izm] - vt[izm]));
          const float oy = ((up[izp] - ut[izp]) - (up[izm] - ut[izm]))
                         - ((wp[ixp] - wt[ixp]) - (wp[ixm] - wt[ixm]));
          const float oz = ((vp[ixp] - vt[ixp]) - (vp[ixm] - vt[ixm]))
                         - ((up[iyp] - ut[iyp]) - (up[iym] - ut[iym]));
          accN += __builtin_sqrtf(ox * ox + oy * oy + oz * oz);
          accC += 1.f;
        }
      }
    }
    __syncthreads();   // all reads of slot (z-1)&3 done before it is reloaded
  }

  // ---- deterministic per-block reduction ----
  red[tid] = accN; red[256 + tid] = accC;
  __syncthreads();
  for (int s = 128; s > 0; s >>= 1) {
    if (tid < s) {
      red[tid]       += red[tid + s];
      red[256 + tid] += red[256 + tid + s];
    }
    __syncthreads();
  }
  if (tid == 0) { ws[2 * bid] = red[0]; ws[2 * bid + 1] = red[256]; }
}

__global__ __launch_bounds__(256) void finalize_kernel(
    const float* __restrict__ ws, float* __restrict__ out, int nblk)
{
  __shared__ float red[512];
  float n = 0.f, c = 0.f;
  for (int i = threadIdx.x; i < nblk; i += 256) {
    n += ws[2 * i];
    c += ws[2 * i + 1];
  }
  red[threadIdx.x] = n; red[256 + threadIdx.x] = c;
  __syncthreads();
  for (int s = 128; s > 0; s >>= 1) {
    if (threadIdx.x < s) {
      red[threadIdx.x]       += red[threadIdx.x + s];
      red[256 + threadIdx.x] += red[256 + threadIdx.x + s];
    }
    __syncthreads();
  }
  if (threadIdx.x == 0) out[0] = red[0] / red[256];
}

extern "C" void kernel_launch(void* const* d_in, const int* in_sizes, int n_in,
                              void* d_out, int out_size, void* d_ws, size_t ws_size,
                              hipStream_t stream) {
  const float* P = (const float*)d_in[0];  // predicts (2,4,64,256,256) f32
  const float* T = (const float*)d_in[1];  // targets  (2,4,64,256,256) f32
  const float* M = (const float*)d_in[2];  // masks    (2,1,64,256,256) f32
  float* ws = (float*)d_ws;                // 2 floats per block (8 KB)

  vortdiff_kernel<<<dim3(NBLK), dim3(256), 0, stream>>>(P, T, M, ws);
  finalize_kernel<<<dim3(1), dim3(256), 0, stream>>>(ws, (float*)d_out, NBLK);
}